// RGCN_61469571940705
// MI455X (gfx1250) — compile-verified
//
#include <hip/hip_runtime.h>

#define N_NODES 60373
#define N_RELS  90
#define H_DIM   16
#define OUT_DIM 4

typedef __attribute__((ext_vector_type(2))) float v2f;
typedef __attribute__((ext_vector_type(8))) float v8f;

// ---------------- utility kernels ----------------

__global__ void zero_kernel(unsigned int* p, int n) {
    int t = blockIdx.x * blockDim.x + threadIdx.x;
    if (t < n) p[t] = 0u;
}

__global__ void hist_kernel(const int* __restrict__ rel, int* __restrict__ bin_count, int E) {
    int t = blockIdx.x * blockDim.x + threadIdx.x;
    if (t < E) atomicAdd(&bin_count[rel[t]], 1);
}

// 90 bins: trivial serial scan on one lane. Also builds the prefix of
// 16-edge group counts per relation and initializes scatter cursors.
__global__ void scan_kernel(const int* __restrict__ bin_count, int* bin_start,
                            int* grp_pref, int* cursor) {
    if (blockIdx.x == 0 && threadIdx.x == 0) {
        int acc = 0, g = 0;
        for (int r = 0; r < N_RELS; ++r) {
            bin_start[r] = acc;
            cursor[r]    = acc;
            grp_pref[r]  = g;
            int c = bin_count[r];
            acc += c;
            g   += (c + 15) >> 4;
        }
        bin_start[N_RELS] = acc;
        grp_pref[N_RELS]  = g;
    }
}

__global__ void scatter_kernel(const int* __restrict__ rel, int* __restrict__ cursor,
                               int* __restrict__ perm, int E) {
    int t = blockIdx.x * blockDim.x + threadIdx.x;
    if (t < E) {
        int p = atomicAdd(&cursor[rel[t]], 1);
        perm[p] = t;
    }
}

// ---------------- layer 1: gather W1 row, scale, scatter-add ----------------
// 4 threads per edge, float4 each -> one coalesced 64B row per edge.
__global__ void layer1_kernel(const float* __restrict__ W1, const float* __restrict__ norm,
                              const int* __restrict__ src, const int* __restrict__ dst,
                              const int* __restrict__ rel, float* __restrict__ h1, int E) {
    int t = blockIdx.x * blockDim.x + threadIdx.x;
    if (t >= 4 * E) return;
    int e = t >> 2, q = t & 3;
    float nm = norm[e];
    const float4 w = *(const float4*)(W1 + ((size_t)rel[e] * N_NODES + src[e]) * H_DIM + q * 4);
    float* o = h1 + (size_t)dst[e] * H_DIM + q * 4;
    atomicAdd(o + 0, w.x * nm);
    atomicAdd(o + 1, w.y * nm);
    atomicAdd(o + 2, w.z * nm);
    atomicAdd(o + 3, w.w * nm);
}

// ---------------- layers 2/3: relation-bucketed WMMA tiles ----------------
// One wave32 handles a group of <=16 edges of one relation.
// A (16x16 f32) = relu(hin)[src_e] rows; B = W[r]; D = A@B via 4x chained
// V_WMMA_F32_16X16X4_F32. Then scale by per-edge norm and atomic-scatter.
//
// Rows m >= cnt simply duplicate the group's first edge (finite data); their
// D rows are never scattered and their norm is forced to 0, so A needs no
// zero-padding selects -> straight-line float2 loads feeding the WMMA chain.
//
// f32 16x16x4 operand layout (ISA 7.12.2): lane L -> m=L%16, up=L/16.
//   A: M=m, K(local) = v + 2*up  (v = VGPR index 0..1)
//   B: N=m, K(local) = v + 2*up
//   C/D: N=m, M = v + 8*up       (v = 0..7)
template <int NCOLS>
__global__ void rgcn_wmma_layer(const float* __restrict__ hin, const float* __restrict__ W,
                                const float* __restrict__ norm, const int* __restrict__ src,
                                const int* __restrict__ dst, const int* __restrict__ perm,
                                const int* __restrict__ bin_start, const int* __restrict__ grp_pref,
                                float* __restrict__ out) {
    int wave = (blockIdx.x * blockDim.x + threadIdx.x) >> 5;
    int lane = threadIdx.x & 31;
    int Gtot = grp_pref[N_RELS];
    if (wave >= Gtot) return;                       // uniform per wave

    // binary search: relation r with grp_pref[r] <= wave < grp_pref[r+1]
    int lo = 0, hi = N_RELS;
    while (hi - lo > 1) {
        int mid = (lo + hi) >> 1;
        if (grp_pref[mid] <= wave) lo = mid; else hi = mid;
    }
    int r    = lo;
    int base = bin_start[r] + ((wave - grp_pref[r]) << 4);
    int cnt  = bin_start[r + 1] - base;
    if (cnt > 16) cnt = 16;

    int m  = lane & 15;
    int up = lane >> 4;
    bool rowok = (m < cnt);

    int   e  = perm[base + (rowok ? m : 0)];        // clamp: stays inside perm
    int   s  = src[e];
    int   d  = dst[e];
    float nm = rowok ? norm[e] : 0.0f;              // kills padded rows on scatter

    const float* hrow = hin + (size_t)s * H_DIM;
    const float* Wr   = W + (size_t)r * H_DIM * NCOLS;

    v8f c = {};
#pragma unroll
    for (int k4 = 0; k4 < 4; ++k4) {
        int kk = k4 * 4 + up * 2;                   // absolute K for v=0
        // one 8-byte load per lane per step (kk is even -> 8B aligned)
        const float2 hv = *(const float2*)(hrow + kk);
        v2f a, b;
        a.x = fmaxf(hv.x, 0.0f);                    // lazy relu
        a.y = fmaxf(hv.y, 0.0f);
        b.x = (NCOLS == H_DIM || m < NCOLS) ? Wr[(kk + 0) * NCOLS + (m & (NCOLS - 1))] : 0.0f;
        b.y = (NCOLS == H_DIM || m < NCOLS) ? Wr[(kk + 1) * NCOLS + (m & (NCOLS - 1))] : 0.0f;
        c = __builtin_amdgcn_wmma_f32_16x16x4_f32(false, a, false, b, (short)0, c,
                                                  false, false);
    }

    // scatter: lane holds D[M = v + 8*up][N = m]
#pragma unroll
    for (int v = 0; v < 8; ++v) {
        int   M  = v + up * 8;
        int   dM = __shfl(d, M, 32);                // lane M holds edge M's dst
        float nM = __shfl(nm, M, 32);
        if (M < cnt && m < NCOLS)
            atomicAdd(&out[(size_t)dM * NCOLS + m], c[v] * nM);
    }
}

// ---------------- softmax over 4 classes ----------------
__global__ void softmax_kernel(const float* __restrict__ logits, float* __restrict__ out, int n) {
    int t = blockIdx.x * blockDim.x + threadIdx.x;
    if (t >= n) return;
    float4 l = *(const float4*)(logits + (size_t)t * 4);
    float mx = fmaxf(fmaxf(l.x, l.y), fmaxf(l.z, l.w));
    float e0 = __expf(l.x - mx), e1 = __expf(l.y - mx);
    float e2 = __expf(l.z - mx), e3 = __expf(l.w - mx);
    float inv = 1.0f / (e0 + e1 + e2 + e3);
    float4 o;
    o.x = e0 * inv; o.y = e1 * inv; o.z = e2 * inv; o.w = e3 * inv;
    *(float4*)(out + (size_t)t * 4) = o;
}

// ---------------- launch ----------------
extern "C" void kernel_launch(void* const* d_in, const int* in_sizes, int n_in,
                              void* d_out, int out_size, void* d_ws, size_t ws_size,
                              hipStream_t stream) {
    const float* W1   = (const float*)d_in[0];
    const float* W2   = (const float*)d_in[1];
    const float* W3   = (const float*)d_in[2];
    const float* norm = (const float*)d_in[3];
    const int*   src  = (const int*)d_in[4];
    const int*   dst  = (const int*)d_in[5];
    const int*   rel  = (const int*)d_in[6];
    const int E = in_sizes[3];

    const size_t N16 = (size_t)N_NODES * H_DIM;
    const size_t N4  = (size_t)N_NODES * OUT_DIM;

    float* h1        = (float*)d_ws;
    float* h2        = h1 + N16;
    float* logits    = h2 + N16;
    int*   bin_count = (int*)(logits + N4);
    int*   bin_start = bin_count + 96;
    int*   grp_pref  = bin_start + 96;
    int*   cursor    = grp_pref + 96;
    int*   perm      = cursor + 96;

    // zero h1, h2, logits, bin_count (contiguous prefix of ws)
    int zero_words = (int)(2 * N16 + N4) + 96;
    zero_kernel<<<(zero_words + 255) / 256, 256, 0, stream>>>((unsigned int*)d_ws, zero_words);

    hist_kernel<<<(E + 255) / 256, 256, 0, stream>>>(rel, bin_count, E);
    scan_kernel<<<1, 32, 0, stream>>>(bin_count, bin_start, grp_pref, cursor);
    scatter_kernel<<<(E + 255) / 256, 256, 0, stream>>>(rel, cursor, perm, E);

    layer1_kernel<<<(4 * E + 255) / 256, 256, 0, stream>>>(W1, norm, src, dst, rel, h1, E);

    int Gmax   = (E + 15) / 16 + N_RELS;   // upper bound on #groups
    int blocks = (Gmax + 3) / 4;           // 128 threads = 4 waves/block
    rgcn_wmma_layer<H_DIM><<<blocks, 128, 0, stream>>>(h1, W2, norm, src, dst, perm,
                                                       bin_start, grp_pref, h2);
    rgcn_wmma_layer<OUT_DIM><<<blocks, 128, 0, stream>>>(h2, W3, norm, src, dst, perm,
                                                         bin_start, grp_pref, logits);

    softmax_kernel<<<(N_NODES + 255) / 256, 256, 0, stream>>>(logits, (float*)d_out, N_NODES);
}